// _AttentionPool_16398185136870
// MI455X (gfx1250) — compile-verified
//
#include <hip/hip_runtime.h>
#include <hip/hip_bf16.h>

typedef float v2f __attribute__((ext_vector_type(2)));
typedef float v8f __attribute__((ext_vector_type(8)));

#define D_MODEL 1024
#define SEQ     8192
#define BATCH   8
#define NEG_BIG (-3.402823466e38f)

// ---------------------------------------------------------------------------
// Kernel 1: scores[b][s] = dot(x[b][s][:], q)  (raw dot, no scale/mask yet)
// One wave -> 16 sequence rows, block = 8 waves -> 128 rows.
// WMMA f32 16x16x4: A = x tile (16 rows x 4 K), B = q broadcast over all 16 N.
// Every column of D holds the 16 scores; lanes with N==0 write them out.
// ---------------------------------------------------------------------------
__global__ void __launch_bounds__(256)
ap_scores_kernel(const float* __restrict__ x,
                 const float* __restrict__ query,
                 float* __restrict__ scores) {
    __shared__ __align__(16) float qs[D_MODEL];

    const int tid  = threadIdx.x;
    const int b    = blockIdx.x >> 6;           // 64 tiles of 128 rows per batch
    const int st   = (blockIdx.x & 63) * 128;   // seq tile base
    const int wave = tid >> 5;
    const int lane = tid & 31;
    const int half = lane >> 4;                 // 0: K=0..1, 1: K=2..3
    const int m    = lane & 15;                 // row within 16-row tile

    // Stage query in LDS: 256 threads x float4 = 1024 floats.
    *(float4*)&qs[tid * 4] = *(const float4*)&query[tid * 4];
    __syncthreads();

    const int row = st + wave * 16 + m;
    const float* xrow = x + ((size_t)b * SEQ + row) * D_MODEL;

    v8f c = {};
    for (int dd = 0; dd < D_MODEL; dd += 64) {
        __builtin_prefetch(xrow + dd + 512, 0, 0);   // global_prefetch_b8
#pragma unroll
        for (int d = dd; d < dd + 64; d += 4) {
            v2f a  = *(const v2f*)(xrow + d + 2 * half);  // A: K = 2*half + {0,1}
            v2f bq = *(const v2f*)(qs   + d + 2 * half);  // B: same K, all N cols
            c = __builtin_amdgcn_wmma_f32_16x16x4_f32(
                    false, a, false, bq, (short)0, c, false, false);
        }
    }

    // D layout: VGPR r, lanes 0-15 -> M=r, lanes 16-31 -> M=8+r, N=lane&15.
    // Lanes with N==0 (lanes 0 and 16) hold all 16 scores between them.
    if (m == 0) {
        float* sp = scores + (size_t)b * SEQ + st + wave * 16 + half * 8;
        float4 lo = {c[0], c[1], c[2], c[3]};
        float4 hi = {c[4], c[5], c[6], c[7]};
        *(float4*)(sp)     = lo;
        *(float4*)(sp + 4) = hi;
    }
}

// ---------------------------------------------------------------------------
// Kernel 2: masked softmax over S per batch -> normalized weights.
// One block (256 threads) per batch; 32 scores per thread; LDS reductions.
// ---------------------------------------------------------------------------
__global__ void __launch_bounds__(256)
ap_softmax_kernel(const float* __restrict__ scores,
                  const unsigned char* __restrict__ mask,  // jnp.bool_ = 1 byte
                  float* __restrict__ weights) {
    __shared__ float red[256];
    const int b   = blockIdx.x;
    const int tid = threadIdx.x;
    const float scale = 0.03125f;  // 1/sqrt(1024)

    float v[SEQ / 256];
    float mx = NEG_BIG;
#pragma unroll
    for (int i = 0; i < SEQ / 256; ++i) {
        const int s = tid + i * 256;
        const float raw = scores[(size_t)b * SEQ + s];
        const bool  pad = mask[(size_t)b * SEQ + s] != 0;
        v[i] = pad ? NEG_BIG : raw * scale;
        mx = fmaxf(mx, v[i]);
    }
    red[tid] = mx;
    __syncthreads();
    for (int o = 128; o > 0; o >>= 1) {
        if (tid < o) red[tid] = fmaxf(red[tid], red[tid + o]);
        __syncthreads();
    }
    const float gmax = red[0];
    __syncthreads();

    float sm = 0.f;
#pragma unroll
    for (int i = 0; i < SEQ / 256; ++i) {
        const float e = (v[i] == NEG_BIG) ? 0.f : expf(v[i] - gmax);
        v[i] = e;
        sm += e;
    }
    red[tid] = sm;
    __syncthreads();
    for (int o = 128; o > 0; o >>= 1) {
        if (tid < o) red[tid] += red[tid + o];
        __syncthreads();
    }
    const float gsum = red[0];
    const float inv  = gsum > 0.f ? 1.f / gsum : 0.f;

#pragma unroll
    for (int i = 0; i < SEQ / 256; ++i)
        weights[(size_t)b * SEQ + tid + i * 256] = v[i] * inv;
}

// ---------------------------------------------------------------------------
// Kernel 3: zero the accumulated output (out is poisoned by the harness).
// ---------------------------------------------------------------------------
__global__ void ap_zero_kernel(float* __restrict__ out, int n) {
    const int i = blockIdx.x * blockDim.x + threadIdx.x;
    if (i < n) out[i] = 0.f;
}

// ---------------------------------------------------------------------------
// Kernel 4: out[b][d] += sum_s w[b][s] * x[b][s][d]
// WMMA f32 16x16x4: A = weights broadcast down M (all rows identical),
// B = 4x16 tile of x (K = 4 seq rows, N = 16 d columns).
// Wave -> 16 d columns, block (8 waves) -> 128 d, grid splits S into chunks
// of 512 accumulated with global_atomic_add_f32.
// Grid: ((b*8 + dchunk)*16 + schunk), 8*8*16 = 1024 blocks.
// ---------------------------------------------------------------------------
__global__ void __launch_bounds__(256)
ap_weighted_sum_kernel(const float* __restrict__ x,
                       const float* __restrict__ weights,
                       float* __restrict__ out) {
    const int sc = blockIdx.x & 15;
    const int dc = (blockIdx.x >> 4) & 7;
    const int b  = blockIdx.x >> 7;

    const int wave = threadIdx.x >> 5;
    const int lane = threadIdx.x & 31;
    const int half = lane >> 4;
    const int n    = lane & 15;

    const int d0    = dc * 128 + wave * 16;
    const int sbase = sc * 512;

    const float* xb = x + (size_t)b * SEQ * D_MODEL;
    const float* wb = weights + (size_t)b * SEQ;

    v8f c = {};
#pragma unroll 4
    for (int s = sbase; s < sbase + 512; s += 4) {
        // A[m][k] = w[s+k] for every m (broadcast down rows).
        v2f aw = *(const v2f*)(wb + s + 2 * half);
        // B[k][n] = x[s+k][d0+n]; per lane: b.x -> K=2*half, b.y -> K=2*half+1.
        v2f bx;
        bx.x = xb[(size_t)(s + 2 * half)     * D_MODEL + d0 + n];
        bx.y = xb[(size_t)(s + 2 * half + 1) * D_MODEL + d0 + n];
        c = __builtin_amdgcn_wmma_f32_16x16x4_f32(
                false, aw, false, bx, (short)0, c, false, false);
    }

    // All D rows identical; lane l (<16) VGPR0 holds the partial for d0+l.
    if (lane < 16)
        atomicAdd(&out[(size_t)b * D_MODEL + d0 + lane], c[0]);
}

// ---------------------------------------------------------------------------
extern "C" void kernel_launch(void* const* d_in, const int* in_sizes, int n_in,
                              void* d_out, int out_size, void* d_ws, size_t ws_size,
                              hipStream_t stream) {
    const float*         x    = (const float*)d_in[0];
    const unsigned char* mask = (const unsigned char*)d_in[1];
    const float*         q    = (const float*)d_in[2];
    float*               out  = (float*)d_out;

    float* scores  = (float*)d_ws;                 // 8*8192 f32 = 256 KB
    float* weights = scores + (size_t)BATCH * SEQ; // 8*8192 f32 = 256 KB

    // 1) raw dot products (WMMA f32)
    ap_scores_kernel<<<BATCH * (SEQ / 128), 256, 0, stream>>>(x, q, scores);
    // 2) masked softmax -> normalized weights
    ap_softmax_kernel<<<BATCH, 256, 0, stream>>>(scores, mask, weights);
    // 3) zero output accumulator
    ap_zero_kernel<<<(BATCH * D_MODEL + 255) / 256, 256, 0, stream>>>(
        out, BATCH * D_MODEL);
    // 4) weighted reduction (WMMA f32 + f32 atomics)
    ap_weighted_sum_kernel<<<BATCH * 8 * 16, 256, 0, stream>>>(x, weights, out);
}